// GridPoolingLayer_5549097747077
// MI455X (gfx1250) — compile-verified
//
#include <hip/hip_runtime.h>

typedef __attribute__((ext_vector_type(1))) float v1f;
typedef __attribute__((ext_vector_type(2))) float v2f;
typedef __attribute__((ext_vector_type(4))) float v4f;
typedef __attribute__((ext_vector_type(8))) float v8f;

template <int VW> struct vecf;
template <> struct vecf<1> { typedef v1f type; };
template <> struct vecf<2> { typedef v2f type; };
template <> struct vecf<4> { typedef v4f type; };

// D = ones(16x4) x B(4x16) + C  => every row of D is the 4-row column sum.
static __device__ __forceinline__ v8f wmma_acc4(v2f b, v8f acc) {
    v2f ones; ones.x = 1.0f; ones.y = 1.0f;
    return __builtin_amdgcn_wmma_f32_16x16x4_f32(
        false, ones, false, b, (short)0, acc, false, false);
}

// ---------------------------------------------------------------------------
// Pass 0: per-index stripe start + length along a masked axis.
// ---------------------------------------------------------------------------
__global__ void __launch_bounds__(256)
seg_kernel(const int* __restrict__ mask, int L,
           int* __restrict__ s0, int* __restrict__ scnt) {
    int i = blockIdx.x * blockDim.x + threadIdx.x;
    if (i >= L) return;
    int m = mask[i];
    int a = i;
    while (a > 0 && mask[a - 1] == m) --a;
    int b = i;
    while (b + 1 < L && mask[b + 1] == m) ++b;
    s0[i]   = a;
    scnt[i] = b - a + 1;
}

// ---------------------------------------------------------------------------
// Stripe-mean along a strided axis. One wave owns 16*VW contiguous minor-axis
// positions: lane L covers columns q..q+VW-1 (q = cbase + VW*(L&15)); tile t
// accumulates column residue t, so each lane moves VW floats per row with one
// b128/b64 load (and stores b128/b64 on broadcast). WMMA columns are
// independent (A = ones), so edge lanes clamp their load address and simply
// don't store -> zero exec juggling in the hot loop. Lanes 0-15 supply
// K=0/1 (B VGPR0/1), lanes 16-31 K=2/3 per the fp32 B layout.
// Host guarantees: VW==4 only when climit % 4 == 0 (keeps clamp aligned).
// ---------------------------------------------------------------------------
template <int VW>
static __device__ __forceinline__ void
stripe_mean(const float* __restrict__ src, float* __restrict__ dst,
            int cnt, size_t stride, int cbase, int climit, int lane) {
    typedef typename vecf<VW>::type vf;
    const int q   = cbase + (lane & 15) * VW;                  // lane's first col
    const int qcl = (q + VW - 1 < climit) ? q
                   : (climit >= VW ? climit - VW : 0);         // clamped for loads
    const int khalf = (lane >> 4) << 1;                        // 0 | 2

    v8f acc[VW];
#pragma unroll
    for (int t = 0; t < VW; ++t) acc[t] = (v8f){};

    const float* p = src + (size_t)khalf * stride + qcl;

    int k = 0;
#pragma unroll 2
    for (; k + 4 <= cnt; k += 4) {                             // full quads, no masks
        vf r0 = *(const vf*)p;                                 // row k+khalf
        vf r1 = *(const vf*)(p + stride);                      // row k+khalf+1
#pragma unroll
        for (int t = 0; t < VW; ++t) {
            v2f b; b.x = r0[t]; b.y = r1[t];
            acc[t] = wmma_acc4(b, acc[t]);
        }
        p += 4 * stride;
    }
    if (k < cnt) {                                             // masked tail (1..3 rows)
        const int k0 = k + khalf;
        vf r0 = {}, r1 = {};
        if (k0     < cnt) r0 = *(const vf*)p;
        if (k0 + 1 < cnt) r1 = *(const vf*)(p + stride);
#pragma unroll
        for (int t = 0; t < VW; ++t) {
            v2f b; b.x = r0[t]; b.y = r1[t];
            acc[t] = wmma_acc4(b, acc[t]);
        }
    }

    const float s = 1.0f / (float)cnt;
    vf mean;
#pragma unroll
    for (int t = 0; t < VW; ++t) mean[t] = acc[t][0] * s;      // cols q..q+VW-1

    float* o = dst + q;
    if (q + VW - 1 < climit) {                                 // wide broadcast store
        for (int r = (lane >> 4); r < cnt; r += 2)             // lanes0-15 even rows,
            *(vf*)(o + (size_t)r * stride) = mean;             // lanes16-31 odd rows
    } else {
#pragma unroll
        for (int t = 0; t < VW; ++t)                           // ragged edge columns
            if (q + t < climit)
                for (int r = (lane >> 4); r < cnt; r += 2)
                    o[(size_t)r * stride + t] = mean[t];
    }
}

// ---------------------------------------------------------------------------
// Pass 1: row-stripe mean over H, minor axis = W*C (contiguous).
// ---------------------------------------------------------------------------
template <int VW>
__global__ void __launch_bounds__(256)
row_pass(const float* __restrict__ x, float* __restrict__ y,
         const int* __restrict__ h0, const int* __restrict__ hcnt,
         int H, int WC) {
    const int h = blockIdx.y;
    if (h0[h] != h) return;                     // stripe representatives only
    const int wave  = threadIdx.x >> 5;
    const int nbase = (blockIdx.x * 8 + wave) * (16 * VW);
    if (nbase >= WC) return;                    // wave-uniform exit
    stripe_mean<VW>(x + (size_t)h * WC, y + (size_t)h * WC,
                    hcnt[h], (size_t)WC, nbase, WC, threadIdx.x & 31);
}

// ---------------------------------------------------------------------------
// Pass 2: col-stripe mean over W (stride C), in place on y. Each block reads
// exactly the elements it rewrites -> race-free.
// ---------------------------------------------------------------------------
template <int VW>
__global__ void __launch_bounds__(256)
col_pass(float* __restrict__ y,
         const int* __restrict__ w0, const int* __restrict__ wcnt,
         int W, int C) {
    const int h = blockIdx.z;
    const int w = blockIdx.y;
    if (w0[w] != w) return;                     // stripe representatives only
    const int wave  = threadIdx.x >> 5;
    const int cbase = (blockIdx.x * 8 + wave) * (16 * VW);
    if (cbase >= C) return;                     // wave-uniform exit
    float* base = y + ((size_t)h * W + w) * C;  // w == stripe start
    stripe_mean<VW>(base, base, wcnt[w], (size_t)C, cbase, C, threadIdx.x & 31);
}

// ---------------------------------------------------------------------------
extern "C" void kernel_launch(void* const* d_in, const int* in_sizes, int n_in,
                              void* d_out, int out_size, void* d_ws, size_t ws_size,
                              hipStream_t stream) {
    const float* x      = (const float*)d_in[0];
    const int*   h_mask = (const int*)d_in[1];
    const int*   v_mask = (const int*)d_in[2];
    float*       out    = (float*)d_out;

    const int H  = in_sizes[1];                 // h_mask is [1, H]
    const int W  = in_sizes[2];                 // v_mask is [1, W]
    const int C  = in_sizes[0] / (H * W);       // input is [1, H, W, C]
    const int WC = W * C;

    // Workspace (ints): h0[H] | hcnt[H] | w0[W] | wcnt[W]  (~8 KB)
    int* wsI  = (int*)d_ws;
    int* h0   = wsI;
    int* hcnt = wsI + H;
    int* w0   = wsI + 2 * H;
    int* wcnt = wsI + 2 * H + W;

    seg_kernel<<<(H + 255) / 256, 256, 0, stream>>>(h_mask, H, h0, hcnt);
    seg_kernel<<<(W + 255) / 256, 256, 0, stream>>>(v_mask, W, w0, wcnt);

    // Vector width: b128 path needs aligned, non-straddling 4-col groups,
    // which holds iff C % 4 == 0 (then WC % 4 == 0 as well).
    const int vw = (C % 4 == 0) ? 4 : (C % 2 == 0) ? 2 : 1;
    const int span = 8 * 16 * vw;               // cols per 256-thread block

    {
        dim3 grid((WC + span - 1) / span, H, 1);
        if (vw == 4)      row_pass<4><<<grid, 256, 0, stream>>>(x, out, h0, hcnt, H, WC);
        else if (vw == 2) row_pass<2><<<grid, 256, 0, stream>>>(x, out, h0, hcnt, H, WC);
        else              row_pass<1><<<grid, 256, 0, stream>>>(x, out, h0, hcnt, H, WC);
    }
    {
        dim3 grid((C + span - 1) / span, W, H);
        if (vw == 4)      col_pass<4><<<grid, 256, 0, stream>>>(out, w0, wcnt, W, C);
        else if (vw == 2) col_pass<2><<<grid, 256, 0, stream>>>(out, w0, wcnt, W, C);
        else              col_pass<1><<<grid, 256, 0, stream>>>(out, w0, wcnt, W, C);
    }
}